// FeatureQuantizer_2130303779235
// MI455X (gfx1250) — compile-verified
//
#include <hip/hip_runtime.h>
#include <hip/hip_bf16.h>

typedef __attribute__((ext_vector_type(2))) float v2f;
typedef __attribute__((ext_vector_type(8))) float v8f;

// ---- output layout (floats), reference return order ----
#define Q_OFF     1LL
#define NQ        (257LL*64*3*256)          // 12,632,064
#define FPERP_OFF (Q_OFF + NQ)              // 12,632,065
#define CPERP_OFF (FPERP_OFF + 1)           // 12,632,066
#define CAVG_OFF  (CPERP_OFF + 1)           // 12,632,067
#define FAVG_OFF  (CAVG_OFF + 1024)         // 12,633,091
#define IDX_OFF   (FAVG_OFF + 8192)         // 12,641,283

// ---- workspace layout (floats) ----
#define WS_W2C 0       // 1024 class codebook squared norms
#define WS_W2F 1024    // 8192 feature codebook squared norms
#define WS_LPC 9216    // 4 class-block loss partials
#define WS_LPF 9220    // 1024 feature-block loss partials

// Init: codebook squared norms, zero counts regions in d_out, zero loss partials.
__global__ __launch_bounds__(256) void init_kernel(const float* __restrict__ cw,
                                                   const float* __restrict__ fw,
                                                   float* __restrict__ ws,
                                                   float* __restrict__ out) {
  int gid = blockIdx.x * 256 + threadIdx.x;
  if (gid < 1024) {
    const float* r = cw + (size_t)gid * 256;
    float s = 0.f;
    for (int c = 0; c < 256; ++c) s += r[c] * r[c];
    ws[WS_W2C + gid] = s;
    out[CAVG_OFF + gid] = 0.f;
  } else if (gid < 9216) {
    int k = gid - 1024;
    const float* r = fw + (size_t)k * 256;
    float s = 0.f;
    for (int c = 0; c < 256; ++c) s += r[c] * r[c];
    ws[WS_W2F + k] = s;
    out[FAVG_OFF + k] = 0.f;
  }
  if (gid < 1028) ws[WS_LPC + gid] = 0.f;
}

// Fused fp32-WMMA distance GEMM + top-3 + gather + loss partials.
// Block = 128 threads (4 waves); block owns 16 rows of X; waves split K.
__global__ __launch_bounds__(128) void vq_topk_kernel(
    const float* __restrict__ X, const float* __restrict__ W,
    const float* __restrict__ w2, float* __restrict__ counts,
    float* __restrict__ qOut, float* __restrict__ idxOut,
    float* __restrict__ lossPartial, int K, int idxOffset) {
  __shared__ float tile[4][16][68];   // padded: row stride 68 floats -> conflict-free column scans
  __shared__ float candV[16][8][3];
  __shared__ int   candI[16][8][3];
  __shared__ int   fIdx[16][3];
  __shared__ float red[128];

  const int tid  = threadIdx.x;
  const int wave = tid >> 5;
  const int lane = tid & 31;
  const int lm   = lane & 15;      // A-row / B-col / C-col within sub-tile
  const int hlf  = lane >> 4;      // lane half
  const int koff = hlf * 2;        // K offset inside the 16x16x4 fragment
  const int m0   = blockIdx.x * 16;

  // running top-3 (ascending) per (row, column-half)
  float t0v = 3.4e38f, t1v = 3.4e38f, t2v = 3.4e38f;
  int   t0i = 0x7fffffff, t1i = 0x7fffffff, t2i = 0x7fffffff;

  const float* xp = X + (size_t)(m0 + lm) * 256 + koff;

  for (int cc = wave * 64; cc < K; cc += 256) {
    const float* wp0 = W + (size_t)(cc +  0 + lm) * 256 + koff;
    const float* wp1 = W + (size_t)(cc + 16 + lm) * 256 + koff;
    const float* wp2 = W + (size_t)(cc + 32 + lm) * 256 + koff;
    const float* wp3 = W + (size_t)(cc + 48 + lm) * 256 + koff;
    v8f acc0 = {0.f,0.f,0.f,0.f,0.f,0.f,0.f,0.f};
    v8f acc1 = {0.f,0.f,0.f,0.f,0.f,0.f,0.f,0.f};
    v8f acc2 = {0.f,0.f,0.f,0.f,0.f,0.f,0.f,0.f};
    v8f acc3 = {0.f,0.f,0.f,0.f,0.f,0.f,0.f,0.f};
#pragma unroll 4
    for (int kk = 0; kk < 256; kk += 4) {
      v2f a  = *(const v2f*)(xp  + kk);
      v2f b0 = *(const v2f*)(wp0 + kk);
      v2f b1 = *(const v2f*)(wp1 + kk);
      v2f b2 = *(const v2f*)(wp2 + kk);
      v2f b3 = *(const v2f*)(wp3 + kk);
      acc0 = __builtin_amdgcn_wmma_f32_16x16x4_f32(false, a, false, b0, (short)0, acc0, false, false);
      acc1 = __builtin_amdgcn_wmma_f32_16x16x4_f32(false, a, false, b1, (short)0, acc1, false, false);
      acc2 = __builtin_amdgcn_wmma_f32_16x16x4_f32(false, a, false, b2, (short)0, acc2, false, false);
      acc3 = __builtin_amdgcn_wmma_f32_16x16x4_f32(false, a, false, b3, (short)0, acc3, false, false);
    }
    // ranking score: ||w||^2 - 2*x.w  (||x||^2 cancels in argmin)
#pragma unroll
    for (int s = 0; s < 4; ++s) {
      float w2v = w2[cc + s * 16 + lm];
      v8f ac = (s == 0) ? acc0 : (s == 1) ? acc1 : (s == 2) ? acc2 : acc3;
#pragma unroll
      for (int v = 0; v < 8; ++v)
        tile[wave][v + 8 * hlf][s * 16 + lm] = w2v - 2.0f * ac[v];
    }
    __syncthreads();
    // each thread scans half a row of this wave's 16x64 tile
#pragma unroll 4
    for (int i = 0; i < 32; ++i) {
      float val = tile[wave][lm][hlf * 32 + i];
      int   idx = cc + hlf * 32 + i;
      if (val < t2v) {
        if (val < t1v) {
          t2v = t1v; t2i = t1i;
          if (val < t0v) { t1v = t0v; t1i = t0i; t0v = val; t0i = idx; }
          else           { t1v = val; t1i = idx; }
        } else { t2v = val; t2i = idx; }
      }
    }
    __syncthreads();
  }

  // merge 8 candidate lists per row
  int slot = wave * 2 + hlf;
  candV[lm][slot][0] = t0v; candV[lm][slot][1] = t1v; candV[lm][slot][2] = t2v;
  candI[lm][slot][0] = t0i; candI[lm][slot][1] = t1i; candI[lm][slot][2] = t2i;
  __syncthreads();

  if (tid < 16) {
    int r = tid;
    unsigned used = 0;
    for (int j = 0; j < 3; ++j) {
      float bv = 3.5e38f; int bi = 0x7fffffff; int bp = 0;
      for (int p = 0; p < 24; ++p) {
        if ((used >> p) & 1u) continue;
        float v  = candV[r][p / 3][p % 3];
        int   id = candI[r][p / 3][p % 3];
        if (v < bv || (v == bv && id < bi)) { bv = v; bi = id; bp = p; }
      }
      used |= 1u << bp;
      fIdx[r][j] = bi;
      idxOut[(size_t)(m0 + r) * 3 + j] = (float)(bi + idxOffset);
      atomicAdd(&counts[bi], 1.0f);  // exact integer-valued float sums: deterministic
    }
  }
  __syncthreads();

  // gather quantized vectors + commitment-loss partial (deterministic block reduction)
  float lsum = 0.f;
  for (int e = tid; e < 16 * 3 * 256; e += 128) {
    int r = e / 768;
    int rem = e - r * 768;
    int j = rem >> 8;
    int c = rem & 255;
    int id = fIdx[r][j];
    float wv = W[(size_t)id * 256 + c];
    float xv = X[(size_t)(m0 + r) * 256 + c];
    qOut[((size_t)(m0 + r) * 3 + j) * 256 + c] = wv;
    float d = wv - xv;
    lsum += d * d;
  }
  red[tid] = lsum;
  __syncthreads();
  for (int o = 64; o; o >>= 1) {
    if (tid < o) red[tid] += red[tid + o];
    __syncthreads();
  }
  if (tid == 0) lossPartial[blockIdx.x] = red[0];
}

// Final scalars: total_loss, avg_probs (in place over counts), perplexities.
__global__ __launch_bounds__(256) void finalize_kernel(const float* __restrict__ ws,
                                                       float* __restrict__ out) {
  __shared__ float rA[256], rB[256];
  int t = threadIdx.x;
  float ac = 0.f, af = 0.f;
  for (int i = t; i < 1028; i += 256) {
    float v = ws[WS_LPC + i];
    if (i < 4) ac += v; else af += v;
  }
  rA[t] = ac; rB[t] = af;
  __syncthreads();
  for (int o = 128; o; o >>= 1) {
    if (t < o) { rA[t] += rA[t + o]; rB[t] += rB[t + o]; }
    __syncthreads();
  }
  if (t == 0)
    out[0] = 0.25f * (rA[0] * (1.0f / 49152.0f)) + 0.25f * (rB[0] * (1.0f / 12582912.0f));
  __syncthreads();

  float s = 0.f;
  for (int k = t; k < 1024; k += 256) {
    float p = out[CAVG_OFF + k] * (1.0f / 64.0f);
    out[CAVG_OFF + k] = p;
    s += p * logf(p + 1e-10f);
  }
  rA[t] = s; __syncthreads();
  for (int o = 128; o; o >>= 1) { if (t < o) rA[t] += rA[t + o]; __syncthreads(); }
  if (t == 0) out[CPERP_OFF] = expf(-rA[0]);
  __syncthreads();

  s = 0.f;
  for (int k = t; k < 8192; k += 256) {
    float p = out[FAVG_OFF + k] * (1.0f / 16384.0f);
    out[FAVG_OFF + k] = p;
    s += p * logf(p + 1e-10f);
  }
  rA[t] = s; __syncthreads();
  for (int o = 128; o; o >>= 1) { if (t < o) rA[t] += rA[t + o]; __syncthreads(); }
  if (t == 0) out[FPERP_OFF] = expf(-rA[0]);
}

extern "C" void kernel_launch(void* const* d_in, const int* in_sizes, int n_in,
                              void* d_out, int out_size, void* d_ws, size_t ws_size,
                              hipStream_t stream) {
  (void)in_sizes; (void)n_in; (void)out_size; (void)ws_size;
  const float* features = (const float*)d_in[0];  // (257, 64, 256)
  const float* cw       = (const float*)d_in[1];  // (1024, 256)
  const float* fw       = (const float*)d_in[2];  // (8192, 256)
  float* out = (float*)d_out;
  float* ws  = (float*)d_ws;

  init_kernel<<<36, 256, 0, stream>>>(cw, fw, ws, out);

  // class VQ: rows 0..63 = features[0]; K=1024
  vq_topk_kernel<<<4, 128, 0, stream>>>(
      features, cw, ws + WS_W2C, out + CAVG_OFF,
      out + Q_OFF, out + IDX_OFF, ws + WS_LPC, 1024, 0);

  // feature VQ: rows = features[1:] flattened (16384 rows); K=8192, idx += 1024
  vq_topk_kernel<<<1024, 128, 0, stream>>>(
      features + 64 * 256, fw, ws + WS_W2F, out + FAVG_OFF,
      out + Q_OFF + 64LL * 3 * 256, out + IDX_OFF + 64LL * 3, ws + WS_LPF, 8192, 1024);

  finalize_kernel<<<1, 256, 0, stream>>>(ws, out);
}